// WTConv2d_62010737820216
// MI455X (gfx1250) — compile-verified
//
#include <hip/hip_runtime.h>
#include <hip/hip_bf16.h>

// ---------------------------------------------------------------------------
// WTConv2d fused kernel for MI455X (gfx1250).
//   - one workgroup per (n,c) plane; whole plane staged in 320KB WGP LDS
//   - x streamed HBM->LDS with the Tensor Data Mover (tensor_load_to_lds),
//     pipelined against the level-0 Haar analysis via s_wait_tensorcnt
//   - base depthwise 3x3 conv executed on the matrix pipe as banded-Toeplitz
//     V_WMMA_F32_16X16X4_F32 accumulation (15 WMMAs per 16x16 tile)
//   - wavelet path algebraically collapsed using the reference's degenerate
//     per-channel filter layout (all 4 subbands of a channel share one 2x2
//     Haar filter dec_base[c>>6]); conv linearity folds the 7 subband convs
//     into two 3x3 convs (U on 56x56, V on 112x112)
// HBM traffic: read x once + write out once = ~410 MB  -> ~17.6 us floor.
// ---------------------------------------------------------------------------

typedef float v2f  __attribute__((ext_vector_type(2)));
typedef float v8f  __attribute__((ext_vector_type(8)));
typedef unsigned int u32x4 __attribute__((ext_vector_type(4)));
typedef int   i32x4 __attribute__((ext_vector_type(4)));
typedef int   i32x8 __attribute__((ext_vector_type(8)));

#define HH   224
#define WW   224
#define HS   112          // H/2
#define QS   56           // H/4
#define NC   256          // channels
#define NTHREADS 512      // 16 wave32s

// LDS layout (float indices)
#define OFF_X 0                       // 224*224 = 50176
#define OFF_S (OFF_X + HH*WW)         // 112*112 = 12544
#define OFF_T (OFF_S + HS*HS)         // 56*56   = 3136
#define OFF_U (OFF_T + QS*QS)         // 56*56   = 3136
#define OFF_V (OFF_U + QS*QS)         // 112*112 = 12544
#define LDS_FLOATS (OFF_V + HS*HS)    // 81536 floats
#define LDS_BYTES  (LDS_FLOATS * 4)   // 326144 B <= 327680 B (320 KB WGP LDS)

// Issue one TDM 2D tile load: nrows x 224 fp32, global -> LDS byte offset.
// D# built per CDNA5 ISA (group0/group1 bitfields).
__device__ __forceinline__ void tdm_load_rows(unsigned lds_byte_off,
                                              const float* gsrc, int nrows) {
  unsigned long long ga = (unsigned long long)(const void*)gsrc;
  u32x4 g0;
  g0.x = 1u;                                            // count=1 (valid), flags 0
  g0.y = lds_byte_off;                                  // lds_addr (bytes)
  g0.z = (unsigned)ga;                                  // global_addr[31:0]
  g0.w = ((unsigned)(ga >> 32) & 0x01FFFFFFu)           // global_addr[56:32]
       | (2u << 30);                                    // type = 2 ("image")
  i32x8 g1;
  g1[0] = (int)(2u << 16);                              // data_size = 2 (4 bytes)
  g1[1] = (int)((unsigned)WW << 16);                    // tensor_dim0[15:0]
  g1[2] = (int)(((unsigned)nrows & 0xFFFFu) << 16);     // tensor_dim1[15:0]
  g1[3] = (int)((unsigned)WW << 16);                    // tile_dim0 = 224
  g1[4] = (int)((unsigned)nrows & 0xFFFFu);             // tile_dim1, tile_dim2=0
  g1[5] = WW;                                           // tensor_dim0_stride lo32
  g1[6] = 0;
  g1[7] = 0;
  i32x4 z4 = {0, 0, 0, 0};                              // groups 2/3 unused (2D)
  i32x8 z8 = {0, 0, 0, 0, 0, 0, 0, 0};                  // extra group (clang-23)
  // clang-23 / therock-10.0 6-arg form:
  __builtin_amdgcn_tensor_load_to_lds(g0, g1, z4, z4, z8, 0);
}

extern "C" __global__ __launch_bounds__(NTHREADS)
void wtconv_fused(const float* __restrict__ x,
                  const float* __restrict__ base_w,
                  const float* __restrict__ base_b,
                  const float* __restrict__ base_scale,
                  const float* __restrict__ wavelet_scale,
                  const float* __restrict__ w_ll,
                  const float* __restrict__ w_lh0,
                  const float* __restrict__ w_hl0,
                  const float* __restrict__ w_hh0,
                  const float* __restrict__ w_lh1,
                  const float* __restrict__ w_hl1,
                  const float* __restrict__ w_hh1,
                  float* __restrict__ out) {
  extern __shared__ float lds[];
  const int plane = blockIdx.x;            // n*256 + c
  const int c    = plane & (NC - 1);
  const int tid  = threadIdx.x;
  const int lane = tid & 31;               // wave32
  const int wave = tid >> 5;               // 16 waves

  const float* xp = x   + (size_t)plane * (HH * WW);
  float*       op = out + (size_t)plane * (HH * WW);

  // Degenerate per-channel Haar filter: fd = c>>6 selects dec/rec kernel.
  // dec[a][b] = 0.25*rowS(a)*colS(b), rec[a][b] = 0.5*rowS(a)*colS(b)
  const int   fd = c >> 6;
  const float dr = (fd & 1) ? -1.f : 1.f;  // sign at row index 1
  const float dc = (fd & 2) ? -1.f : 1.f;  // sign at col index 1

  // Uniform per-channel weights (conv linearity folds the subband convs)
  float bw[9], ws0[9], ws1[9];
#pragma unroll
  for (int t = 0; t < 9; ++t) {
    bw[t]  = base_w[c * 9 + t];
    ws0[t] = w_lh0[c * 9 + t] + w_hl0[c * 9 + t] + w_hh0[c * 9 + t];
    ws1[t] = w_ll [c * 9 + t] + w_lh1[c * 9 + t] + w_hl1[c * 9 + t] + w_hh1[c * 9 + t];
  }
  const float bb  = base_b[c];
  const float bsc = base_scale[c];
  const float wsc = wavelet_scale[c];

  // ---- stage 0: queue all 7 TDM chunk loads (32 rows each) on wave 0 ------
  if (wave == 0) {
#pragma unroll
    for (int k = 0; k < 7; ++k)
      tdm_load_rows((unsigned)((OFF_X + k * 32 * WW) * 4), xp + k * 32 * WW, 32);
  }

  // ---- stage 1: S = stride-2 Haar analysis of X, pipelined on TENSORcnt ---
  const float d00 = 0.25f, d01 = 0.25f * dc, d10 = 0.25f * dr, d11 = 0.25f * dr * dc;
  for (int k = 0; k < 7; ++k) {
    switch (k) {  // chunks complete in order; wait until chunk k has landed
      case 0: __builtin_amdgcn_s_wait_tensorcnt(6); break;
      case 1: __builtin_amdgcn_s_wait_tensorcnt(5); break;
      case 2: __builtin_amdgcn_s_wait_tensorcnt(4); break;
      case 3: __builtin_amdgcn_s_wait_tensorcnt(3); break;
      case 4: __builtin_amdgcn_s_wait_tensorcnt(2); break;
      case 5: __builtin_amdgcn_s_wait_tensorcnt(1); break;
      default: __builtin_amdgcn_s_wait_tensorcnt(0); break;
    }
    __syncthreads();
    for (int p = tid; p < 16 * HS; p += NTHREADS) {
      int i = k * 16 + (p / HS);
      int j = p % HS;
      const float* r0 = lds + OFF_X + (2 * i) * WW + 2 * j;
      lds[OFF_S + i * HS + j] = d00 * r0[0] + d01 * r0[1] + d10 * r0[WW] + d11 * r0[WW + 1];
    }
  }
  __syncthreads();

  // ---- stage 2: T = stride-2 Haar analysis of S ---------------------------
  for (int p = tid; p < QS * QS; p += NTHREADS) {
    int i = p / QS, j = p % QS;
    const float* r0 = lds + OFF_S + (2 * i) * HS + 2 * j;
    lds[OFF_T + p] = d00 * r0[0] + d01 * r0[1] + d10 * r0[HS] + d11 * r0[HS + 1];
  }
  __syncthreads();

  // ---- stage 3: U = conv3x3(T, ws1), zero pad (56x56) ---------------------
  for (int p = tid; p < QS * QS; p += NTHREADS) {
    int i = p / QS, j = p % QS;
    float acc = 0.f;
#pragma unroll
    for (int a = 0; a < 3; ++a) {
      int ii = i + a - 1;
      if (ii < 0 || ii >= QS) continue;
#pragma unroll
      for (int b = 0; b < 3; ++b) {
        int jj = j + b - 1;
        if (jj >= 0 && jj < QS) acc += ws1[a * 3 + b] * lds[OFF_T + ii * QS + jj];
      }
    }
    lds[OFF_U + p] = acc;
  }
  // ---- stage 4: V = conv3x3(S, ws0), zero pad (112x112); S is stable ------
  for (int p = tid; p < HS * HS; p += NTHREADS) {
    int i = p / HS, j = p % HS;
    float acc = 0.f;
#pragma unroll
    for (int a = 0; a < 3; ++a) {
      int ii = i + a - 1;
      if (ii < 0 || ii >= HS) continue;
#pragma unroll
      for (int b = 0; b < 3; ++b) {
        int jj = j + b - 1;
        if (jj >= 0 && jj < HS) acc += ws0[a * 3 + b] * lds[OFF_S + ii * HS + jj];
      }
    }
    lds[OFF_V + p] = acc;
  }
  __syncthreads();

  // ---- stage 5: base 3x3 depthwise conv on the matrix pipe + fused epilog -
  // y_tile(16x16) = sum_{a=0..2} A_a(16x20) * B_a(20x16), B_a banded Toeplitz
  // of the 3 row-a filter taps; K padded 18->20 => 5 K-chunks => 15 WMMAs.
  // B fragments are tile-invariant: build once, hold in VGPRs.
  v2f Bf[15];
  {
    const int n    = lane & 15;
    const int koff = (lane >> 4) * 2;  // VGPR0 holds K=4q+0 / 4q+2 halves
#pragma unroll
    for (int a = 0; a < 3; ++a) {
#pragma unroll
      for (int q = 0; q < 5; ++q) {
        int k0 = 4 * q + koff;
        int b0 = k0 - n, b1 = k0 + 1 - n;   // filter tap index = k - n
        v2f bf;
        bf.x = (b0 == 0) ? bw[a * 3] : (b0 == 1) ? bw[a * 3 + 1] : (b0 == 2) ? bw[a * 3 + 2] : 0.f;
        bf.y = (b1 == 0) ? bw[a * 3] : (b1 == 1) ? bw[a * 3 + 1] : (b1 == 2) ? bw[a * 3 + 2] : 0.f;
        Bf[a * 5 + q] = bf;
      }
    }
  }

  for (int u = wave; u < 14 * 14; u += 16) {   // 196 16x16 tiles, EXEC all-1s
    const int i0 = (u / 14) * 16;
    const int j0 = (u % 14) * 16;
    const int m    = lane & 15;
    const int koff = (lane >> 4) * 2;
    v8f acc = {};
#pragma unroll
    for (int a = 0; a < 3; ++a) {
      const int row = i0 + m + a - 1;
      const bool rok = (row >= 0) && (row < HH);
      const float* xr = lds + OFF_X + row * WW;
#pragma unroll
      for (int q = 0; q < 5; ++q) {
        const int col0 = j0 + 4 * q + koff - 1;
        v2f A;
        A.x = (rok && col0     >= 0 && col0     < WW) ? xr[col0]     : 0.f;
        A.y = (rok && col0 + 1 >= 0 && col0 + 1 < WW) ? xr[col0 + 1] : 0.f;
        acc = __builtin_amdgcn_wmma_f32_16x16x4_f32(
                  false, A, false, Bf[a * 5 + q], (short)0, acc, false, false);
      }
    }
    // D fragment: VGPR r <-> output row i0+r+8*half, lane&15 <-> column j0+n.
    const int n    = lane & 15;
    const int half = lane >> 4;
#pragma unroll
    for (int r = 0; r < 8; ++r) {
      const int oi = i0 + r + 8 * half;
      const int oj = j0 + n;
      const int I = oi >> 1, J = oj >> 1;
      const float Vv  = lds[OFF_V + I * HS + J];
      const float Uv  = lds[OFF_U + (I >> 1) * QS + (J >> 1)];
      const float rIJ = 0.5f * ((I  & 1) ? dr : 1.f) * ((J  & 1) ? dc : 1.f);
      const float rab = 0.5f * ((oi & 1) ? dr : 1.f) * ((oj & 1) ? dc : 1.f);
      const float wt  = rab * (rIJ * Uv + Vv);
      op[oi * WW + oj] = (acc[r] + bb) * bsc + wsc * wt;
    }
  }
}

extern "C" void kernel_launch(void* const* d_in, const int* in_sizes, int n_in,
                              void* d_out, int out_size, void* d_ws, size_t ws_size,
                              hipStream_t stream) {
  (void)in_sizes; (void)n_in; (void)out_size; (void)d_ws; (void)ws_size;
  const float* x   = (const float*)d_in[0];
  const float* bwt = (const float*)d_in[1];
  const float* bb  = (const float*)d_in[2];
  const float* bs  = (const float*)d_in[3];
  const float* wsl = (const float*)d_in[4];
  const float* wll = (const float*)d_in[5];
  const float* wlh0= (const float*)d_in[6];
  const float* whl0= (const float*)d_in[7];
  const float* whh0= (const float*)d_in[8];
  const float* wlh1= (const float*)d_in[9];
  const float* whl1= (const float*)d_in[10];
  const float* whh1= (const float*)d_in[11];
  float* out = (float*)d_out;

  // 326,144 B of dynamic LDS per workgroup (within 320 KB WGP LDS).
  (void)hipFuncSetAttribute((const void*)wtconv_fused,
                            hipFuncAttributeMaxDynamicSharedMemorySize, LDS_BYTES);
  dim3 grid(4 * NC);       // one workgroup per (n, c) plane
  dim3 block(NTHREADS);    // 16 wave32s
  wtconv_fused<<<grid, block, LDS_BYTES, stream>>>(
      x, bwt, bb, bs, wsl, wll, wlh0, whl0, whh0, wlh1, whl1, whh1, out);
}